// ComponentWiseSpline_29265907155470
// MI455X (gfx1250) — compile-verified
//
#include <hip/hip_runtime.h>
#include <cstdint>

// Problem constants (match reference)
#define B_ROWS   65536
#define D_DIM    128
#define K_BINS   8
#define BOUNDF   3.0f
#define MIN_BW   1e-3f
#define MIN_BH   1e-3f
#define MIN_DV   1e-3f
#define EPSF     1e-6f

// Precomputed-table layout in d_ws / LDS (float offsets), SoA: field[k][128].
// Row stride 128 floats = multiple of 64 banks -> bin-indexed gathers are
// conflict-free for d = lane + 32j.
#define OFF_CWE 0       // interior knots + EPS (k=1..7 at rows 0..6): 7*128
#define OFF_CW  896     // cumwidths  (k=0..7): 8*128
#define OFF_IWI 1920    // 1/width    (k=0..7): 8*128
#define OFF_CH  2944    // cumheights (k=0..7): 8*128
#define OFF_H   3968    // heights    (k=0..7): 8*128
#define OFF_DL  4992    // delta=h/w  (k=0..7): 8*128
#define OFF_DR  6016    // derivatives(k=0..8): 9*128
#define TAB_FLOATS 7168
#define TAB_VEC4   1792   // 16-byte chunks; 1792 = 7 * 256 (divergence-free)

#define WAVES_PER_BLOCK 8
#define ROWS_PER_WAVE   4
#define GRID_BLOCKS     (B_ROWS / (WAVES_PER_BLOCK * ROWS_PER_WAVE))  // 2048

typedef __attribute__((ext_vector_type(2))) float v2f;
typedef __attribute__((ext_vector_type(8))) float v8f;

__device__ __forceinline__ float fast_rcp(float v) {
#if defined(__has_builtin) && __has_builtin(__builtin_amdgcn_rcpf)
  return __builtin_amdgcn_rcpf(v);   // v_rcp_f32
#else
  return 1.0f / v;
#endif
}

__device__ __forceinline__ float clamp3(float v) {
#if defined(__has_builtin) && __has_builtin(__builtin_amdgcn_fmed3f)
  return __builtin_amdgcn_fmed3f(v, -BOUNDF, BOUNDF);  // single v_med3
#else
  return fminf(fmaxf(v, -BOUNDF), BOUNDF);
#endif
}

// xor-lane shuffle with immediate pattern: ds_swizzle_b32, group-of-32 mode,
// offset = (xor_mask << 10) | (or_mask << 5) | and_mask(0x1f). No index VALU.
template <int XORMASK>
__device__ __forceinline__ float swz_xor(float v) {
#if defined(__has_builtin) && __has_builtin(__builtin_amdgcn_ds_swizzle)
  return __int_as_float(
      __builtin_amdgcn_ds_swizzle(__float_as_int(v), (XORMASK << 10) | 0x1f));
#else
  return __shfl_xor(v, XORMASK, 32);
#endif
}

// ---------------------------------------------------------------------------
// Kernel 1: build spline tables from [D,K] parameters. 1 block, 1 thread/dim.
// ---------------------------------------------------------------------------
__global__ __launch_bounds__(128) void spline_tables_kernel(
    const float* __restrict__ uw, const float* __restrict__ uh,
    const float* __restrict__ ud, float* __restrict__ tab)
{
  const int d = threadIdx.x;
  if (d >= D_DIM) return;

  float wv[K_BINS], hv[K_BINS];
  float mw = -1e30f, mh = -1e30f;
  #pragma unroll
  for (int k = 0; k < K_BINS; ++k) {
    wv[k] = uw[d * K_BINS + k]; mw = fmaxf(mw, wv[k]);
    hv[k] = uh[d * K_BINS + k]; mh = fmaxf(mh, hv[k]);
  }
  float sw = 0.0f, sh = 0.0f;
  #pragma unroll
  for (int k = 0; k < K_BINS; ++k) {
    wv[k] = __expf(wv[k] - mw); sw += wv[k];
    hv[k] = __expf(hv[k] - mh); sh += hv[k];
  }

  // knots: cumsum of (min + (1-min*K)*softmax), scaled to [-3,3], endpoints forced
  float cw[K_BINS + 1], ch[K_BINS + 1];
  cw[0] = -BOUNDF; ch[0] = -BOUNDF;
  const float scale = 1.0f - MIN_BW * (float)K_BINS;
  float accw = 0.0f, acch = 0.0f;
  #pragma unroll
  for (int k = 0; k < K_BINS; ++k) {
    accw += MIN_BW + scale * (wv[k] / sw);
    acch += MIN_BH + scale * (hv[k] / sh);
    cw[k + 1] = 2.0f * BOUNDF * accw - BOUNDF;
    ch[k + 1] = 2.0f * BOUNDF * acch - BOUNDF;
  }
  cw[K_BINS] = BOUNDF; ch[K_BINS] = BOUNDF;

  #pragma unroll
  for (int k = 1; k < K_BINS; ++k)            // interior knots for the search
    tab[OFF_CWE + (k - 1) * D_DIM + d] = cw[k] + EPSF;
  #pragma unroll
  for (int k = 0; k < K_BINS; ++k) {
    const float wk = cw[k + 1] - cw[k];
    const float hk = ch[k + 1] - ch[k];
    tab[OFF_CW  + k * D_DIM + d] = cw[k];
    tab[OFF_IWI + k * D_DIM + d] = 1.0f / wk;  // exact divide, once
    tab[OFF_CH  + k * D_DIM + d] = ch[k];
    tab[OFF_H   + k * D_DIM + d] = hk;
    tab[OFF_DL  + k * D_DIM + d] = hk / wk;
  }

  // derivatives: pad(softplus(ud)+min, 1 each side with 1-min)
  tab[OFF_DR + 0 * D_DIM + d]      = 1.0f - MIN_DV;
  tab[OFF_DR + K_BINS * D_DIM + d] = 1.0f - MIN_DV;
  #pragma unroll
  for (int k = 1; k < K_BINS; ++k) {
    const float u  = ud[d * (K_BINS - 1) + (k - 1)];
    const float sp = (u > 20.0f) ? u : log1pf(__expf(u));
    tab[OFF_DR + k * D_DIM + d] = MIN_DV + sp;
  }
}

// ---------------------------------------------------------------------------
// Kernel 2: streaming spline evaluation. One wave per row, 4 rows per wave.
// Tables staged to LDS with gfx1250 async global->LDS loads (ASYNCcnt path).
// ---------------------------------------------------------------------------
__global__ __launch_bounds__(256) void spline_fwd_kernel(
    const float* __restrict__ x, const float* __restrict__ tabg,
    float* __restrict__ outU, float* __restrict__ outL)
{
  __shared__ __align__(16) float tab[TAB_FLOATS];
  const int tid = threadIdx.x;

  // ---- async stage 28 KB of tables into LDS (16B per lane per op) ----
  #pragma unroll 1
  for (int i = tid; i < TAB_VEC4; i += 256) {
    const unsigned loff = (unsigned)(uintptr_t)(&tab[4 * i]);   // LDS byte offset
    const unsigned long long g = (unsigned long long)(tabg + 4 * i);
    asm volatile("global_load_async_to_lds_b128 %0, %1, off"
                 :: "v"(loff), "v"(g) : "memory");
  }
#if defined(__has_builtin) && __has_builtin(__builtin_amdgcn_s_wait_asynccnt)
  __builtin_amdgcn_s_wait_asynccnt(0);
#else
  asm volatile("s_wait_asynccnt 0" ::: "memory");
#endif
  __syncthreads();

  const int wave = tid >> 5;
  const int lane = tid & 31;
  const int baseRow = (blockIdx.x * WAVES_PER_BLOCK + wave) * ROWS_PER_WAVE;

  // Each lane owns dims d = lane + 32j (j=0..3) for all of its rows:
  // hoist the 7 interior search knots per dim into registers (28 VGPRs).
  float cwe[4][7];
  #pragma unroll
  for (int j = 0; j < 4; ++j) {
    const int d = lane + 32 * j;
    #pragma unroll
    for (int k = 0; k < 7; ++k)
      cwe[j][k] = tab[OFF_CWE + k * D_DIM + d];
  }

  for (int r = 0; r < ROWS_PER_WAVE; ++r) {
    const int row = baseRow + r;
    const float* xrow = x + (size_t)row * D_DIM;
    float* urow = outU + (size_t)row * D_DIM;
    float partial = 0.0f;

    #pragma unroll
    for (int j = 0; j < 4; ++j) {
      const int d = lane + 32 * j;
      const float xs = xrow[d];
      const float xc = clamp3(xs);             // one v_med3

      // bin = #{interior knots <= xc} : exactly clip(count-1, 0, K-1)
      int bin = 0;
      #pragma unroll
      for (int k = 0; k < 7; ++k)
        bin += (xc >= cwe[j][k]) ? 1 : 0;

      // conflict-free bin-indexed gathers (row stride = 2 x 64 banks,
      // all 7 share one address VGPR via DS immediate offsets)
      const float icw = tab[OFF_CW  + bin * D_DIM + d];
      const float iwi = tab[OFF_IWI + bin * D_DIM + d];
      const float ich = tab[OFF_CH  + bin * D_DIM + d];
      const float ih  = tab[OFF_H   + bin * D_DIM + d];
      const float dl  = tab[OFF_DL  + bin * D_DIM + d];
      const float d0  = tab[OFF_DR  + bin * D_DIM + d];
      const float d1  = tab[OFF_DR  + (bin + 1) * D_DIM + d];

      const float theta = (xc - icw) * iwi;
      const float omt   = 1.0f - theta;
      const float t1m   = theta * omt;
      const float th2   = theta * theta;
      const float num   = ih * (dl * th2 + d0 * t1m);
      const float den   = dl + (d0 + d1 - 2.0f * dl) * t1m;
      const float rden  = fast_rcp(den);
      const float o     = ich + num * rden;
      const float dnum  = dl * dl * (d1 * th2 + 2.0f * dl * t1m + d0 * omt * omt);
      // log(dnum) - 2*log(den) = ln2 * log2(dnum * rden^2) : one v_log_f32
      const float lad   = 0.69314718056f * __log2f(dnum * rden * rden);

      const bool inside = (xs == xc);          // false for NaN and |x|>3
      urow[d]  = inside ? o : xs;
      partial += inside ? lad : 0.0f;
    }

    // ---- wave reduction of log-det partials (imm-pattern ds_swizzle) ----
    float p = partial;
    p += swz_xor<1>(p);
    p += swz_xor<2>(p);
    p += swz_xor<4>(p);
    p += swz_xor<8>(p);
    // Lanes 0-15 hold the low-half sum, lanes 16-31 the high-half sum.
#if defined(__has_builtin) && __has_builtin(__builtin_amdgcn_wmma_f32_16x16x4_f32)
    // A = ones (16x4), B = (p, 0): D[0][n] = b0[n] + b0[n+16] = total row sum.
    v2f a; a[0] = 1.0f; a[1] = 1.0f;
    v2f b; b[0] = p;    b[1] = 0.0f;
    v8f c = {};
    v8f dres = __builtin_amdgcn_wmma_f32_16x16x4_f32(
        /*neg_a=*/false, a, /*neg_b=*/false, b,
        /*c_mod=*/(short)0, c, /*reuse_a=*/false, /*reuse_b=*/false);
    const float tot = dres[0];
#else
    const float tot = p + swz_xor<16>(p);
#endif
    if (lane == 0) outL[row] = tot;
  }
}

// ---------------------------------------------------------------------------
extern "C" void kernel_launch(void* const* d_in, const int* in_sizes, int n_in,
                              void* d_out, int out_size, void* d_ws, size_t ws_size,
                              hipStream_t stream)
{
  (void)in_sizes; (void)n_in; (void)out_size; (void)ws_size;
  const float* x  = (const float*)d_in[0];
  const float* uw = (const float*)d_in[1];
  const float* uh = (const float*)d_in[2];
  const float* ud = (const float*)d_in[3];
  float* tab  = (float*)d_ws;                      // 7168 floats = 28 KB
  float* outU = (float*)d_out;                     // [B, D]
  float* outL = outU + (size_t)B_ROWS * D_DIM;     // [B]

  spline_tables_kernel<<<1, 128, 0, stream>>>(uw, uh, ud, tab);
  spline_fwd_kernel<<<GRID_BLOCKS, 256, 0, stream>>>(x, tab, outU, outL);
}